// GCNConv_20083267076568
// MI455X (gfx1250) — compile-verified
//
#include <hip/hip_runtime.h>
#include <hip/hip_bf16.h>

typedef __attribute__((ext_vector_type(2))) float v2f;
typedef __attribute__((ext_vector_type(8))) float v8f;

#define D_FEAT 128

// ---------------------------------------------------------------------------
// Kernel 1: deg[n] = 1.0 (self-loop contribution)
// ---------------------------------------------------------------------------
__global__ void gcn_deg_init(float* __restrict__ deg, int N) {
    int n = blockIdx.x * blockDim.x + threadIdx.x;
    if (n < N) deg[n] = 1.0f;
}

// ---------------------------------------------------------------------------
// Kernel 2: deg[dst[e]] += 1 for every edge (hardware f32 atomic)
// ---------------------------------------------------------------------------
__global__ void gcn_degree(const long long* __restrict__ dst,
                           float* __restrict__ deg, int E) {
    int e = blockIdx.x * blockDim.x + threadIdx.x;
    if (e < E) unsafeAtomicAdd(&deg[(int)dst[e]], 1.0f);
}

// ---------------------------------------------------------------------------
// Kernel 3: dinv[n] = rsqrt(deg[n])  (deg >= 1 always)
// ---------------------------------------------------------------------------
__global__ void gcn_dinv(const float* __restrict__ deg,
                         float* __restrict__ dinv, int N) {
    int n = blockIdx.x * blockDim.x + threadIdx.x;
    if (n < N) dinv[n] = __frsqrt_rn(deg[n]);
}

// ---------------------------------------------------------------------------
// Kernel 4: agg[n][:] = dinv[n]^2 * x[n][:]   (self-loop term, also zero-init)
// ---------------------------------------------------------------------------
__global__ void gcn_selfloop(const float* __restrict__ x,
                             const float* __restrict__ dinv,
                             float* __restrict__ agg, int N) {
    int i = blockIdx.x * blockDim.x + threadIdx.x;   // over N*128 elements
    if (i < N * D_FEAT) {
        int n = i >> 7;
        float di = dinv[n];
        agg[i] = di * di * x[i];
    }
}

// ---------------------------------------------------------------------------
// Kernel 5: edge scatter. One wave32 per edge; each lane owns 4 contiguous
// floats (float4 coalesced gather of x[src], 4x global_atomic_add_f32 to
// agg[dst]). All data resident in the 192MB L2 -> L2-atomic bound.
// ---------------------------------------------------------------------------
__global__ void gcn_scatter(const float* __restrict__ x,
                            const long long* __restrict__ src,
                            const long long* __restrict__ dst,
                            const float* __restrict__ dinv,
                            float* __restrict__ agg, int E) {
    int gid  = blockIdx.x * blockDim.x + threadIdx.x;
    int e    = gid >> 5;
    int lane = threadIdx.x & 31;
    if (e >= E) return;
    int s = (int)src[e];
    int t = (int)dst[e];
    float c = dinv[s] * dinv[t];
    const float4 xs = *(const float4*)(x + (size_t)s * D_FEAT + lane * 4);
    float* p = agg + (size_t)t * D_FEAT + lane * 4;
    unsafeAtomicAdd(p + 0, c * xs.x);
    unsafeAtomicAdd(p + 1, c * xs.y);
    unsafeAtomicAdd(p + 2, c * xs.z);
    unsafeAtomicAdd(p + 3, c * xs.w);
}

// ---------------------------------------------------------------------------
// Kernel 6: out = agg @ W^T + bias via V_WMMA_F32_16X16X4_F32 (fp32 precision).
// One wave per 16x16 output tile; K-loop of 32 WMMAs over D=128.
//
// 32-bit A 16x4 layout: lanes 0-15 -> M=0..15, VGPR0=K+0, VGPR1=K+1;
//                        lanes 16-31 -> M=0..15, VGPR0=K+2, VGPR1=K+3.
// B (4x16) mirrors with N in place of M. C/D: VGPR v -> M=v (+8 for hi half),
// N = lane&15.
// ---------------------------------------------------------------------------
__global__ void gcn_gemm_wmma(const float* __restrict__ agg,
                              const float* __restrict__ W,     // [128][128] row-major [o][k]
                              const float* __restrict__ bias,
                              float* __restrict__ out, int N) {
    int wavesPerBlock = blockDim.x >> 5;
    int wave = blockIdx.x * wavesPerBlock + (threadIdx.x >> 5);
    int lane = threadIdx.x & 31;

    int row_tile = wave >> 3;        // M tiles (16 rows each)
    int col_tile = wave & 7;         // 8 column tiles of 16 (D_OUT=128)
    int row0 = row_tile << 4;
    if (row0 >= N) return;           // wave-uniform: EXEC stays all-1s for WMMA

    int col0 = col_tile << 4;
    int half = lane >> 4;            // 0: K+{0,1}, 1: K+{2,3}
    int l    = lane & 15;

    int ra = row0 + l;               // A-matrix row for this lane
    if (ra >= N) ra = N - 1;         // safe clamp (result rows guarded at store)
    const float* arow = agg + (size_t)ra * D_FEAT + half * 2;
    const float* brow = W   + (size_t)(col0 + l) * D_FEAT + half * 2;

    v8f c = {};
#pragma unroll
    for (int k = 0; k < D_FEAT; k += 4) {
        v2f a = *(const v2f*)(arow + k);
        v2f b = *(const v2f*)(brow + k);
        // 8 args: (neg_a, A, neg_b, B, c_mod, C, reuse_a, reuse_b)
        c = __builtin_amdgcn_wmma_f32_16x16x4_f32(false, a, false, b,
                                                  (short)0, c, false, false);
    }

    float bv = bias[col0 + l];
#pragma unroll
    for (int v = 0; v < 8; ++v) {
        int m = row0 + v + half * 8;
        if (m < N) out[(size_t)m * D_FEAT + col0 + l] = c[v] + bv;
    }
}

// ---------------------------------------------------------------------------
extern "C" void kernel_launch(void* const* d_in, const int* in_sizes, int n_in,
                              void* d_out, int out_size, void* d_ws, size_t ws_size,
                              hipStream_t stream) {
    const float*     x    = (const float*)d_in[0];
    const long long* ei   = (const long long*)d_in[1];   // int64 [2][E]
    const float*     W    = (const float*)d_in[2];       // [128][128]
    const float*     bias = (const float*)d_in[3];       // [128]
    float*           out  = (float*)d_out;

    const int N = in_sizes[0] / D_FEAT;          // 50000
    const int E = in_sizes[1] / 2;               // 625000
    const long long* src = ei;
    const long long* dst = ei + E;

    // Workspace layout: deg | dinv | agg
    size_t npad = ((size_t)N + 63) & ~(size_t)63;
    float* deg  = (float*)d_ws;
    float* dinv = deg + npad;
    float* agg  = dinv + npad;
    (void)ws_size; (void)n_in; (void)out_size;

    const int BT = 256;
    gcn_deg_init<<<(N + BT - 1) / BT, BT, 0, stream>>>(deg, N);
    gcn_degree  <<<(E + BT - 1) / BT, BT, 0, stream>>>(dst, deg, E);
    gcn_dinv    <<<(N + BT - 1) / BT, BT, 0, stream>>>(deg, dinv, N);
    gcn_selfloop<<<((size_t)N * D_FEAT + BT - 1) / BT, BT, 0, stream>>>(x, dinv, agg, N);

    // one wave per edge
    size_t scatterThreads = (size_t)E * 32;
    gcn_scatter <<<(scatterThreads + BT - 1) / BT, BT, 0, stream>>>(x, src, dst, dinv, agg, E);

    // one wave per 16x16 tile: tiles = ceil(N/16) * 8, 8 waves per block
    int mTiles = (N + 15) / 16;
    int tiles  = mTiles * 8;
    int blocks = (tiles + 7) / 8;
    gcn_gemm_wmma<<<blocks, BT, 0, stream>>>(agg, W, bias, out, N);
}